// TripletLossFilter_9337258902043
// MI455X (gfx1250) — compile-verified
//
#include <hip/hip_runtime.h>
#include <math.h>

typedef float v2f __attribute__((ext_vector_type(2)));
typedef float v8f __attribute__((ext_vector_type(8)));

#define N_ROWS   8192
#define K_DIM    256
#define N_TILES  (N_ROWS / 16)            // 512 tiles of 16 rows
#define N_PAIRS  (N_TILES / 2)            // 256 blocks of 32 rows (2 i-tiles/wave)
#define N_CHUNKS 8                        // j-tiles split into 8 chunks
#define JT_PER_CHUNK (N_TILES / N_CHUNKS) // 64 j-tiles per chunk
#define MARGIN   0.3f
#define EPS      1e-12f

// -------- Kernel A: per-row squared norms --------------------------------
__global__ __launch_bounds__(256)
void sq_kernel(const float* __restrict__ x, float* __restrict__ sq) {
    const int i = blockIdx.x * 256 + threadIdx.x;      // row id, grid = 32 blocks
    const float4* row = (const float4*)(x + (size_t)i * K_DIM);
    float s = 0.0f;
#pragma unroll 8
    for (int k = 0; k < K_DIM / 4; ++k) {
        float4 v = row[k];
        s += v.x * v.x + v.y * v.y + v.z * v.z + v.w * v.w;
    }
    sq[i] = s;
}

// -------- Kernel B: fused WMMA Gram tiles + hardest pos/neg reduction -----
// One wave (32 threads) per block, TWO 16-row i-tiles register-blocked:
// each B fragment read from L2 feeds two WMMAs (one per i-tile chain).
// blockIdx.x = i-tile pair, blockIdx.y = j-chunk.
__global__ __launch_bounds__(32)
void tile_kernel(const float* __restrict__ x,
                 const int*   __restrict__ t32,       // int64 labels, read low dword
                 const float* __restrict__ sq,
                 float* __restrict__ ap_part,         // [N_CHUNKS][N_ROWS]
                 float* __restrict__ an_part) {       // [N_CHUNKS][N_ROWS]
    const int it0   = blockIdx.x * 2;     // first i-tile
    const int it1   = it0 + 1;            // second i-tile
    const int chunk = blockIdx.y;
    const int lane  = threadIdx.x;
    const int half  = lane >> 4;
    const int nidx  = lane & 15;

    // ---- Preload A fragments for both i-tiles (2 x 16x256) --------------
    // A layout (f32 16x4): lane L holds M = L&15, K = kk*4 + 2*(L>>4) + {0,1}
    const float* arow0 = x + (size_t)(it0 * 16 + nidx) * K_DIM + 2 * half;
    const float* arow1 = x + (size_t)(it1 * 16 + nidx) * K_DIM + 2 * half;
    v2f afragA[64], afragB[64];
#pragma unroll
    for (int kk = 0; kk < 64; ++kk) {
        afragA[kk] = *(const v2f*)(arow0 + kk * 4);
        afragB[kk] = *(const v2f*)(arow1 + kk * 4);
    }

    // Per-lane row metadata for the D layout: VGPR r <-> row M = r + 8*half
    float sqiA[8], sqiB[8];
    int   tiA[8],  tiB[8];
#pragma unroll
    for (int r = 0; r < 8; ++r) {
        const int rowA = it0 * 16 + r + 8 * half;
        const int rowB = it1 * 16 + r + 8 * half;
        sqiA[r] = sq[rowA];  tiA[r] = t32[2 * rowA];
        sqiB[r] = sq[rowB];  tiB[r] = t32[2 * rowB];
    }

    float apA[8], anA[8], apB[8], anB[8];
#pragma unroll
    for (int r = 0; r < 8; ++r) {
        apA[r] = -__builtin_inff(); anA[r] = __builtin_inff();
        apB[r] = -__builtin_inff(); anB[r] = __builtin_inff();
    }

    const int jt0 = chunk * JT_PER_CHUNK;
    for (int j = 0; j < JT_PER_CHUNK; ++j) {
        const int jt = jt0 + j;
        // B layout (f32 4x16): lane L holds N = L&15, K = kk*4 + 2*(L>>4) + {0,1}
        const float* brow = x + (size_t)(jt * 16 + nidx) * K_DIM + 2 * half;

        const float sqj = sq[jt * 16 + nidx];
        const int   tj  = t32[2 * (jt * 16 + nidx)];

        v8f accA = {}, accB = {};

        // Double-buffered B fragments, groups of 8; each fragment drives
        // two WMMAs (i-tile A chain + i-tile B chain).
        v2f bbuf[2][8];
#pragma unroll
        for (int t = 0; t < 8; ++t)
            bbuf[0][t] = *(const v2f*)(brow + t * 4);

#pragma unroll
        for (int g = 0; g < 8; ++g) {
            const int cur = g & 1;
            if (g < 7) {
#pragma unroll
                for (int t = 0; t < 8; ++t)
                    bbuf[cur ^ 1][t] = *(const v2f*)(brow + ((g + 1) * 8 + t) * 4);
            }
#pragma unroll
            for (int t = 0; t < 8; ++t) {
                const int kk = g * 8 + t;
                accA = __builtin_amdgcn_wmma_f32_16x16x4_f32(
                    false, afragA[kk], false, bbuf[cur][t], (short)0, accA, false, false);
                accB = __builtin_amdgcn_wmma_f32_16x16x4_f32(
                    false, afragB[kk], false, bbuf[cur][t], (short)0, accB, false, false);
            }
        }

        // Epilogue: distances + masked running max/min for both i-tiles
#pragma unroll
        for (int r = 0; r < 8; ++r) {
            float dA = sqrtf(fmaxf(sqiA[r] + sqj - 2.0f * accA[r], EPS));
            float dB = sqrtf(fmaxf(sqiB[r] + sqj - 2.0f * accB[r], EPS));
            bool sA = (tiA[r] == tj);
            bool sB = (tiB[r] == tj);
            apA[r] = sA ? fmaxf(apA[r], dA) : apA[r];
            anA[r] = sA ? anA[r] : fminf(anA[r], dA);
            apB[r] = sB ? fmaxf(apB[r], dB) : apB[r];
            anB[r] = sB ? anB[r] : fminf(anB[r], dB);
        }
    }

    // ---- Cross-lane reduce within each 16-lane half (rows r+8*half) -----
#pragma unroll
    for (int r = 0; r < 8; ++r) {
#pragma unroll
        for (int m = 1; m < 16; m <<= 1) {
            apA[r] = fmaxf(apA[r], __shfl_xor(apA[r], m, 32));
            anA[r] = fminf(anA[r], __shfl_xor(anA[r], m, 32));
            apB[r] = fmaxf(apB[r], __shfl_xor(apB[r], m, 32));
            anB[r] = fminf(anB[r], __shfl_xor(anB[r], m, 32));
        }
    }

    if (nidx == 0) {   // lane 0 writes rows 0..7, lane 16 writes rows 8..15
        const int baseA = chunk * N_ROWS + it0 * 16 + 8 * half;
        const int baseB = chunk * N_ROWS + it1 * 16 + 8 * half;
#pragma unroll
        for (int r = 0; r < 8; ++r) {
            ap_part[baseA + r] = apA[r];
            an_part[baseA + r] = anA[r];
            ap_part[baseB + r] = apB[r];
            an_part[baseB + r] = anB[r];
        }
    }
}

// -------- Kernel C: combine chunks, hinge, per-block partial sums --------
__global__ __launch_bounds__(256)
void hinge_kernel(const float* __restrict__ ap_part,
                  const float* __restrict__ an_part,
                  float* __restrict__ partials) {
    __shared__ float sm[256];
    const int i = blockIdx.x * 256 + threadIdx.x;      // row id, grid = 32
    float ap = -__builtin_inff(), an = __builtin_inff();
#pragma unroll
    for (int c = 0; c < N_CHUNKS; ++c) {
        ap = fmaxf(ap, ap_part[c * N_ROWS + i]);
        an = fminf(an, an_part[c * N_ROWS + i]);
    }
    sm[threadIdx.x] = fmaxf(ap - an + MARGIN, 0.0f);
    __syncthreads();
    for (int s = 128; s > 0; s >>= 1) {
        if (threadIdx.x < s) sm[threadIdx.x] += sm[threadIdx.x + s];
        __syncthreads();
    }
    if (threadIdx.x == 0) partials[blockIdx.x] = sm[0];
}

// -------- Kernel D: deterministic final sum -> mean -----------------------
__global__ void final_kernel(const float* __restrict__ partials, float* __restrict__ out) {
    if (threadIdx.x == 0) {
        float s = 0.0f;
        for (int b = 0; b < 32; ++b) s += partials[b];
        out[0] = s / (float)N_ROWS;
    }
}

// -------- Launch ----------------------------------------------------------
extern "C" void kernel_launch(void* const* d_in, const int* in_sizes, int n_in,
                              void* d_out, int out_size, void* d_ws, size_t ws_size,
                              hipStream_t stream) {
    const float* x   = (const float*)d_in[0];
    const int*   t32 = (const int*)d_in[1];   // int64 targets; low dwords hold values

    float* ws       = (float*)d_ws;
    float* sq       = ws;                               // 8192
    float* ap_part  = ws + N_ROWS;                      // 8 * 8192
    float* an_part  = ap_part + N_CHUNKS * N_ROWS;      // 8 * 8192
    float* partials = an_part + N_CHUNKS * N_ROWS;      // 32

    sq_kernel<<<N_ROWS / 256, 256, 0, stream>>>(x, sq);
    tile_kernel<<<dim3(N_PAIRS, N_CHUNKS), 32, 0, stream>>>(x, t32, sq, ap_part, an_part);
    hinge_kernel<<<N_ROWS / 256, 256, 0, stream>>>(ap_part, an_part, partials);
    final_kernel<<<1, 32, 0, stream>>>(partials, (float*)d_out);
}